// MeshLoss_84293028151620
// MI455X (gfx1250) — compile-verified
//
#include <hip/hip_runtime.h>

// MeshLoss: loss = mean((net-fem)^2) + 0.1 * (1/12) * sum(dx^2+dy^2+dz^2) over pred interior.
// Chamfer outputs in the reference are dead code (not used in the returned loss).
//
// Shapes: net/fem/pred = (4,3,32,32,32) f32 flat = 393216; output = 1 f32 scalar.
// Memory-bound (~4.7MB); WMMA f32 16x16x4 with an all-ones B matrix is used as the
// accumulation engine: C += A x ones sums all 64 A elements into the C tile.

typedef __attribute__((ext_vector_type(2))) float v2f;
typedef __attribute__((ext_vector_type(8))) float v8f;

#define N_MESH 393216          // 4*3*32*32*32
#define W_FEM  (1.0f / 393216.0f)
#define W_REG  (0.1f / 12.0f)
#define NBLK   256
#define NTHR   256
#define WAVES_PER_BLK (NTHR / 32)

__device__ __forceinline__ float wave_reduce_add(float v) {
#pragma unroll
    for (int m = 16; m >= 1; m >>= 1) v += __shfl_xor(v, m, 32);
    return v;
}

__global__ __launch_bounds__(NTHR) void meshloss_partials(
    const float* __restrict__ net, const float* __restrict__ fem,
    const float* __restrict__ pred, float* __restrict__ ws) {
    const int lane   = threadIdx.x & 31;
    const int wave   = threadIdx.x >> 5;
    const int gwave  = blockIdx.x * WAVES_PER_BLK + wave;
    const int nwaves = gridDim.x * WAVES_PER_BLK;
    const int stride = nwaves * 64;            // 64 f32 consumed per wave per WMMA

    v2f ones; ones.x = 1.0f; ones.y = 1.0f;    // B matrix: every element 1.0
    v8f cf = {};                               // fem-MSE accumulator tile
    v8f cr = {};                               // regularization accumulator tile

    // ---- Phase A: sum (net[i]-fem[i])^2, N divisible by 64, coalesced ----
    for (int i = gwave * 64; i < N_MESH; i += stride) {
        int j0 = i + lane, j1 = i + 32 + lane;
        float d0 = net[j0] - fem[j0];
        float d1 = net[j1] - fem[j1];
        v2f a; a.x = d0 * d0; a.y = d1 * d1;
        cf = __builtin_amdgcn_wmma_f32_16x16x4_f32(false, a, false, ones,
                                                   (short)0, cf, false, false);
    }

    // ---- Phase B: sum dx^2+dy^2+dz^2 over pred; padded 32^3 domain == pred layout.
    // Boundary faces contribute exactly 0 via offset select (branch-free, EXEC stays full).
    for (int i = gwave * 64; i < N_MESH; i += stride) {
        v2f a;
#pragma unroll
        for (int h = 0; h < 2; ++h) {
            int j = i + lane + h * 32;
            int z = j & 31, y = (j >> 5) & 31, x = (j >> 10) & 31;
            float base = pred[j];
            float dx = pred[j + ((x < 31) ? 1024 : 0)] - base;
            float dy = pred[j + ((y < 31) ? 32   : 0)] - base;
            float dz = pred[j + ((z < 31) ? 1    : 0)] - base;
            float v = dx * dx + dy * dy + dz * dz;
            if (h == 0) a.x = v; else a.y = v;
        }
        cr = __builtin_amdgcn_wmma_f32_16x16x4_f32(false, a, false, ones,
                                                   (short)0, cr, false, false);
    }

    // C tile holds 16 identical columns of row-sums: Sum(all 256 slots) = 16 * total.
    float p = 0.0f;
#pragma unroll
    for (int v = 0; v < 8; ++v) p += cf[v] * W_FEM + cr[v] * W_REG;
    p = wave_reduce_add(p) * (1.0f / 16.0f);

    __shared__ float smem[WAVES_PER_BLK];
    if (lane == 0) smem[wave] = p;
    __syncthreads();
    if (threadIdx.x == 0) {
        float s = 0.0f;
#pragma unroll
        for (int w = 0; w < WAVES_PER_BLK; ++w) s += smem[w];
        ws[blockIdx.x] = s;
    }
}

__global__ __launch_bounds__(32) void meshloss_final(
    const float* __restrict__ ws, float* __restrict__ out, int n) {
    float s = 0.0f;
    for (int i = (int)threadIdx.x; i < n; i += 32) s += ws[i];
    s = wave_reduce_add(s);
    if (threadIdx.x == 0) out[0] = s;
}

extern "C" void kernel_launch(void* const* d_in, const int* in_sizes, int n_in,
                              void* d_out, int out_size, void* d_ws, size_t ws_size,
                              hipStream_t stream) {
    (void)in_sizes; (void)n_in; (void)out_size; (void)ws_size;
    const float* net  = (const float*)d_in[0];  // network_mesh
    // d_in[1] = pc : dead code in the reference loss (chamfer results unused)
    const float* fem  = (const float*)d_in[2];  // fem_mesh
    const float* pred = (const float*)d_in[3];  // pred
    float* ws  = (float*)d_ws;
    float* out = (float*)d_out;

    meshloss_partials<<<NBLK, NTHR, 0, stream>>>(net, fem, pred, ws);
    meshloss_final<<<1, 32, 0, stream>>>(ws, out, NBLK);
}